// ConvTemporalGraphical_44573170598205
// MI455X (gfx1250) — compile-verified
//
#include <hip/hip_runtime.h>
#include <hip/hip_bf16.h>
#include <stdint.h>

typedef float v2f __attribute__((ext_vector_type(2)));
typedef float v8f __attribute__((ext_vector_type(8)));
typedef int   v4i __attribute__((ext_vector_type(4)));

#define AS1 __attribute__((address_space(1)))
#define AS3 __attribute__((address_space(3)))

#if defined(__HIP_DEVICE_COMPILE__) && __has_builtin(__builtin_amdgcn_global_load_async_to_lds_b128)
#define HAVE_ASYNC_LDS 1
#else
#define HAVE_ASYNC_LDS 0
#endif

// Problem constants
static constexpr int Nb = 32;            // batch
static constexpr int Cc = 64;            // channels
static constexpr int Tt = 128;           // time
static constexpr int Vv = 25;            // vertices
static constexpr int Ff = Cc * Tt * Vv;  // 204800 flattened features
static constexpr int Hh = 256;           // hidden
static constexpr int Ee = 4;             // experts

// Split-K GEMM1 config
static constexpr int NCHUNK = 400;       // K chunks
static constexpr int KC = 512;           // K per chunk
static constexpr int SUB = 128;          // K per LDS stage (double-buffered)
static constexpr int NSUB = KC / SUB;    // 4 stages per chunk
static constexpr int LSTRIDE = 132;      // LDS row stride (floats), conflict padding
static constexpr int MSTRIDE = 260;      // LDS row stride for the 256-wide MLP tile

// ws float-offset layout
static constexpr size_t PART_SZ = (size_t)NCHUNK * Nb * Hh;  // 3,276,800 floats
static constexpr size_t H1_OFF  = PART_SZ;
static constexpr size_t WSM_OFF = PART_SZ + (size_t)Nb * Hh;

__device__ __forceinline__ float eluf(float v) {
  return v > 0.f ? v : (__expf(v) - 1.f);
}

__device__ __forceinline__ v8f wmma4(v2f a, v2f b, v8f c) {
  // V_WMMA_F32_16X16X4_F32 : D = A(16x4,f32) * B(4x16,f32) + C(16x16,f32)
  return __builtin_amdgcn_wmma_f32_16x16x4_f32(
      /*neg_a=*/false, a, /*neg_b=*/false, b,
      /*c_mod=*/(short)0, c, /*reuse_a=*/false, /*reuse_b=*/false);
}

__device__ __forceinline__ void wait_async_all() {
#if HAVE_ASYNC_LDS
  asm volatile("s_wait_asynccnt 0x0" ::: "memory");
#endif
}

// ---------------------------------------------------------------------------
// Kernel 1: split-K GEMM  P_chunk[m,n] = sum_{k in chunk} x[m,k] * W0[k,n]
// grid = NCHUNK blocks, 128 threads (4 waves). Wave w owns N cols [64w,64w+64).
// x staged in LDS via async-to-LDS double buffering; W0 streamed NT once.
// ---------------------------------------------------------------------------
__global__ void __launch_bounds__(128) k_gemm1(const float* __restrict__ x,
                                               const float* __restrict__ W0,
                                               float* __restrict__ part) {
  __shared__ float Xl[2][Nb * LSTRIDE];
  const int tid = threadIdx.x;
  const int wid = tid >> 5;
  const int lane = tid & 31;
  const int lh = lane >> 4;   // 0|1 (K sub-pair select)
  const int ll = lane & 15;

  const int chunk = blockIdx.x;
  const int kbase = chunk * KC;

  // Stage x[0:32, ksub : ksub+SUB) into Xl[buf]. 32 rows * 32 float4 = 1024
  // transfers over 128 threads = 8 per thread; each wave covers a full row
  // (512B contiguous) per instruction.
  auto stage = [&](int sub, int buf) {
    const int ksub = kbase + sub * SUB;
#pragma unroll
    for (int it = 0; it < 8; ++it) {
      const int i = tid + it * 128;
      const int m = i >> 5;        // row
      const int c4 = i & 31;       // float4 within row
      const float* gp = x + (size_t)m * Ff + ksub + c4 * 4;
      float* lp = &Xl[buf][m * LSTRIDE + c4 * 4];
#if HAVE_ASYNC_LDS
      __builtin_amdgcn_global_load_async_to_lds_b128(
          (AS1 v4i*)gp, (AS3 v4i*)lp, 0, 0);
#else
      *(float4*)lp = *(const float4*)gp;
#endif
    }
  };

  v8f acc[2][4];
  const v8f z8 = {0.f, 0.f, 0.f, 0.f, 0.f, 0.f, 0.f, 0.f};
#pragma unroll
  for (int i = 0; i < 2; ++i)
#pragma unroll
    for (int j = 0; j < 4; ++j) acc[i][j] = z8;

  stage(0, 0);
  wait_async_all();
  __syncthreads();

  for (int sub = 0; sub < NSUB; ++sub) {
    const int buf = sub & 1;
    if (sub + 1 < NSUB) stage(sub + 1, buf ^ 1);  // overlap fill with compute

    const int ksub = kbase + sub * SUB;
    for (int ks = 0; ks < SUB; ks += 4) {
      const int kk = ks + 2 * lh;          // local K for this lane's pair
      const v2f a0 = *(const v2f*)(&Xl[buf][ll * LSTRIDE + kk]);
      const v2f a1 = *(const v2f*)(&Xl[buf][(ll + 16) * LSTRIDE + kk]);
      const int kg = ksub + kk;            // global K
#pragma unroll
      for (int nt = 0; nt < 4; ++nt) {
        const int n = wid * 64 + nt * 16 + ll;
        v2f b;
        b.x = __builtin_nontemporal_load(&W0[(size_t)kg * Hh + n]);
        b.y = __builtin_nontemporal_load(&W0[(size_t)(kg + 1) * Hh + n]);
        acc[0][nt] = wmma4(a0, b, acc[0][nt]);
        acc[1][nt] = wmma4(a1, b, acc[1][nt]);
      }
    }
    // Own async loads must complete before barrier-signal so the next stage's
    // data is visible to all waves after barrier-wait.
    wait_async_all();
    __syncthreads();
  }

  // Write deterministic per-chunk partials (C/D layout: M = r + 8*lh, N = ll)
  float* pd = part + (size_t)chunk * (Nb * Hh);
#pragma unroll
  for (int mt = 0; mt < 2; ++mt)
#pragma unroll
    for (int nt = 0; nt < 4; ++nt)
#pragma unroll
      for (int r = 0; r < 8; ++r) {
        const int m = mt * 16 + r + 8 * lh;
        const int n = wid * 64 + nt * 16 + ll;
        pd[m * Hh + n] = acc[mt][nt][r];
      }
}

// ---------------------------------------------------------------------------
// Kernel 2a: fixed-order reduction over NCHUNK partials + bias + ELU -> h1
// ---------------------------------------------------------------------------
__global__ void __launch_bounds__(256) k_reduce_elu(const float* __restrict__ part,
                                                    const float* __restrict__ b0,
                                                    float* __restrict__ h1) {
  const int idx = blockIdx.x * 256 + threadIdx.x;  // [0, 8192)
  float s = 0.f;
  for (int ch = 0; ch < NCHUNK; ++ch) s += part[(size_t)ch * (Nb * Hh) + idx];
  s += b0[idx & (Hh - 1)];
  h1[idx] = eluf(s);
}

// ---------------------------------------------------------------------------
// Kernel 2b: h2 = elu(h1 @ W1 + b1)  (WMMA), logits = h2 @ W2 + b2, softmax.
// Single block, 128 threads (4 waves).
// ---------------------------------------------------------------------------
__global__ void __launch_bounds__(128) k_mlp_softmax(const float* __restrict__ h1,
                                                     const float* __restrict__ W1,
                                                     const float* __restrict__ b1,
                                                     const float* __restrict__ W2,
                                                     const float* __restrict__ b2,
                                                     float* __restrict__ wsm) {
  __shared__ float Xl[Nb * MSTRIDE];
  __shared__ float Lg[Nb * Ee];
  const int tid = threadIdx.x;
  const int wid = tid >> 5;
  const int lane = tid & 31;
  const int lh = lane >> 4;
  const int ll = lane & 15;

  for (int i = tid; i < Nb * Hh; i += 128) {
    const int m = i >> 8;
    const int n = i & 255;
    Xl[m * MSTRIDE + n] = h1[i];
  }
  __syncthreads();

  v8f acc[2][4];
  const v8f z8 = {0.f, 0.f, 0.f, 0.f, 0.f, 0.f, 0.f, 0.f};
#pragma unroll
  for (int i = 0; i < 2; ++i)
#pragma unroll
    for (int j = 0; j < 4; ++j) acc[i][j] = z8;

  for (int ks = 0; ks < Hh; ks += 4) {
    const int kk = ks + 2 * lh;
    const v2f a0 = *(const v2f*)(&Xl[ll * MSTRIDE + kk]);
    const v2f a1 = *(const v2f*)(&Xl[(ll + 16) * MSTRIDE + kk]);
#pragma unroll
    for (int nt = 0; nt < 4; ++nt) {
      const int n = wid * 64 + nt * 16 + ll;
      v2f b;
      b.x = W1[(size_t)kk * Hh + n];
      b.y = W1[(size_t)(kk + 1) * Hh + n];
      acc[0][nt] = wmma4(a0, b, acc[0][nt]);
      acc[1][nt] = wmma4(a1, b, acc[1][nt]);
    }
  }
  __syncthreads();  // everyone done reading Xl; reuse it for h2

#pragma unroll
  for (int mt = 0; mt < 2; ++mt)
#pragma unroll
    for (int nt = 0; nt < 4; ++nt)
#pragma unroll
      for (int r = 0; r < 8; ++r) {
        const int m = mt * 16 + r + 8 * lh;
        const int n = wid * 64 + nt * 16 + ll;
        Xl[m * MSTRIDE + n] = eluf(acc[mt][nt][r] + b1[n]);
      }
  __syncthreads();

  // logits: 128 threads = 32 rows x 4 experts
  {
    const int m = tid >> 2;
    const int e = tid & 3;
    float s = b2[e];
    for (int k = 0; k < Hh; ++k) s += Xl[m * MSTRIDE + k] * W2[k * Ee + e];
    Lg[m * Ee + e] = s;
  }
  __syncthreads();

  if (tid < Nb) {
    const float l0 = Lg[tid * 4 + 0], l1 = Lg[tid * 4 + 1];
    const float l2 = Lg[tid * 4 + 2], l3 = Lg[tid * 4 + 3];
    const float mx = fmaxf(fmaxf(l0, l1), fmaxf(l2, l3));
    const float e0 = __expf(l0 - mx), e1 = __expf(l1 - mx);
    const float e2 = __expf(l2 - mx), e3 = __expf(l3 - mx);
    const float inv = 1.f / (e0 + e1 + e2 + e3);
    wsm[tid * 4 + 0] = e0 * inv;
    wsm[tid * 4 + 1] = e1 * inv;
    wsm[tid * 4 + 2] = e2 * inv;
    wsm[tid * 4 + 3] = e3 * inv;
  }
}

// ---------------------------------------------------------------------------
// Kernel 3: per (n,t): AS = sum_e w[n,e]*A[e,t,:,:]; out = X_nt(64x25) @ AS(25x25)
// WMMA with K padded 25->28, N padded 25->32 (2 tiles). 4096 blocks, 4 waves.
// ---------------------------------------------------------------------------
__global__ void __launch_bounds__(128) k_graphconv(const float* __restrict__ x,
                                                   const float* __restrict__ A,
                                                   const float* __restrict__ wsm,
                                                   float* __restrict__ out) {
  __shared__ float ASl[28 * 33];  // [Kpad=28][Npad=32 (+1 pad)]
  __shared__ float Xl2[64 * 28];  // [C=64][Kpad=28]
  const int tid = threadIdx.x;
  const int wid = tid >> 5;
  const int lane = tid & 31;
  const int lh = lane >> 4;
  const int ll = lane & 15;
  const int bid = blockIdx.x;
  const int n = bid / Tt;
  const int t = bid % Tt;

  for (int i = tid; i < 28 * 33; i += 128) ASl[i] = 0.f;
  for (int i = tid; i < 64 * 28; i += 128) Xl2[i] = 0.f;
  __syncthreads();

  const float g0 = wsm[n * 4 + 0], g1 = wsm[n * 4 + 1];
  const float g2 = wsm[n * 4 + 2], g3 = wsm[n * 4 + 3];
  for (int i = tid; i < Vv * Vv; i += 128) {
    const int v = i / Vv;
    const int w = i - v * Vv;
    const float* Ap = A + (size_t)t * (Vv * Vv) + i;
    const float s = g0 * Ap[0] +
                    g1 * Ap[(size_t)Tt * Vv * Vv] +
                    g2 * Ap[2 * (size_t)Tt * Vv * Vv] +
                    g3 * Ap[3 * (size_t)Tt * Vv * Vv];
    ASl[v * 33 + w] = s;
  }
  for (int i = tid; i < Cc * Vv; i += 128) {
    const int c = i / Vv;
    const int v = i - c * Vv;
    Xl2[c * 28 + v] = x[(size_t)n * Ff + (size_t)c * (Tt * Vv) + t * Vv + v];
  }
  __syncthreads();

  v8f acc0 = {0.f, 0.f, 0.f, 0.f, 0.f, 0.f, 0.f, 0.f};
  v8f acc1 = acc0;
#pragma unroll
  for (int kb = 0; kb < 28; kb += 4) {
    const int kk = kb + 2 * lh;
    const v2f a = *(const v2f*)(&Xl2[(wid * 16 + ll) * 28 + kk]);
    v2f b0v, b1v;
    b0v.x = ASl[kk * 33 + ll];
    b0v.y = ASl[(kk + 1) * 33 + ll];
    b1v.x = ASl[kk * 33 + 16 + ll];
    b1v.y = ASl[(kk + 1) * 33 + 16 + ll];
    acc0 = wmma4(a, b0v, acc0);
    acc1 = wmma4(a, b1v, acc1);
  }

  float* op = out + (size_t)n * Ff + t * Vv;
#pragma unroll
  for (int r = 0; r < 8; ++r) {
    const int m = r + 8 * lh;
    const int c = wid * 16 + m;
    op[(size_t)c * (Tt * Vv) + ll] = acc0[r];
    const int w2c = 16 + ll;
    if (w2c < Vv) op[(size_t)c * (Tt * Vv) + w2c] = acc1[r];
  }
}

// ---------------------------------------------------------------------------
extern "C" void kernel_launch(void* const* d_in, const int* in_sizes, int n_in,
                              void* d_out, int out_size, void* d_ws, size_t ws_size,
                              hipStream_t stream) {
  (void)in_sizes; (void)n_in; (void)out_size; (void)ws_size;
  const float* x  = (const float*)d_in[0];
  const float* W0 = (const float*)d_in[1];
  const float* b0 = (const float*)d_in[2];
  const float* W1 = (const float*)d_in[3];
  const float* b1 = (const float*)d_in[4];
  const float* W2 = (const float*)d_in[5];
  const float* b2 = (const float*)d_in[6];
  const float* A  = (const float*)d_in[7];
  float* out = (float*)d_out;

  float* wsf  = (float*)d_ws;
  float* part = wsf;             // NCHUNK * 32 * 256 floats (~13.1 MB)
  float* h1   = wsf + H1_OFF;    // 8192 floats
  float* wsm  = wsf + WSM_OFF;   // 128 floats

  hipLaunchKernelGGL(k_gemm1, dim3(NCHUNK), dim3(128), 0, stream, x, W0, part);
  hipLaunchKernelGGL(k_reduce_elu, dim3(Nb * Hh / 256), dim3(256), 0, stream, part, b0, h1);
  hipLaunchKernelGGL(k_mlp_softmax, dim3(1), dim3(128), 0, stream, h1, W1, b1, W2, b2, wsm);
  hipLaunchKernelGGL(k_graphconv, dim3(Nb * Tt), dim3(128), 0, stream, x, A, wsm, out);
}